// FilterDetections_40535901340296
// MI455X (gfx1250) — compile-verified
//
#include <hip/hip_runtime.h>
#include <stdint.h>

#define BDIM   256      // 8 wave32 waves
#define NBINS  2048
#define KTOP   300
#define CAP    4096
#define SORT_T 1024
#define HBLK   128      // histogram blocks per batch
#define CBLK   256      // compaction blocks per batch

// ---- CDNA5 async global->LDS helpers (ASYNCcnt-tracked DMA path) ----------
__device__ __forceinline__ uint32_t lds_off(const void* p) {
    // generic pointer to LDS = {shared aperture hi32, lds offset lo32}
    return (uint32_t)(uintptr_t)p;
}
__device__ __forceinline__ void async_load_b128(uint32_t ldsaddr, const float* gaddr) {
    asm volatile("global_load_async_to_lds_b128 %0, %1, off"
                 :: "v"(ldsaddr), "v"(gaddr)
                 : "memory");
}
#define WAIT_ASYNCCNT(n) asm volatile("s_wait_asynccnt " #n ::: "memory")

__device__ __forceinline__ int score_bin(float s) {
    int bin = (int)((s - 0.5f) * (float)(2 * NBINS));
    bin = bin < 0 ? 0 : bin;
    return bin > (NBINS - 1) ? (NBINS - 1) : bin;
}

// ---- kernel 1: zero histogram + counters ----------------------------------
__global__ void zero_ws_kernel(int* __restrict__ p, int n) {
    int i = blockIdx.x * blockDim.x + threadIdx.x;
    if (i < n) p[i] = 0;
}

// ---- kernel 2: histogram of scores > 0.5, async-to-LDS double buffered ----
__global__ __launch_bounds__(BDIM)
void hist_kernel(const float* __restrict__ cls, int* __restrict__ hist,
                 int nPerBatch, int tilesPerBatch) {
    __shared__ int lh[NBINS];
    __shared__ __align__(16) float sbuf[2][BDIM * 4];

    const int tid = threadIdx.x;
    const int b   = blockIdx.y;
    const int bx  = blockIdx.x;
    const int nblk = gridDim.x;

    for (int i = tid; i < NBINS; i += BDIM) lh[i] = 0;
    __syncthreads();

    const float* base = cls + (size_t)b * (size_t)nPerBatch;

    int nt = 0;
    for (int t = bx; t < tilesPerBatch; t += nblk) nt++;

    const uint32_t l0 = lds_off(&sbuf[0][tid * 4]);
    const uint32_t l1 = lds_off(&sbuf[1][tid * 4]);

    if (nt > 0)
        async_load_b128(l0, base + (size_t)bx * (BDIM * 4) + tid * 4);

    int t = bx;
    for (int i = 0; i < nt; ++i, t += nblk) {
        if (i + 1 < nt) {
            // prefetch next tile into the other buffer, then wait for oldest
            async_load_b128((i & 1) ? l0 : l1,
                            base + (size_t)(t + nblk) * (BDIM * 4) + tid * 4);
            WAIT_ASYNCCNT(1);
        } else {
            WAIT_ASYNCCNT(0);
        }
        __syncthreads();                       // tile i visible to all waves
        float4 v = *(const float4*)&sbuf[i & 1][tid * 4];
        if (v.x > 0.5f) atomicAdd(&lh[score_bin(v.x)], 1);
        if (v.y > 0.5f) atomicAdd(&lh[score_bin(v.y)], 1);
        if (v.z > 0.5f) atomicAdd(&lh[score_bin(v.z)], 1);
        if (v.w > 0.5f) atomicAdd(&lh[score_bin(v.w)], 1);
        __syncthreads();                       // done reading before re-issue
    }

    // tail elements (nPerBatch % (BDIM*4)) handled by block 0, plain loads
    if (bx == 0) {
        for (int i = tilesPerBatch * (BDIM * 4) + tid; i < nPerBatch; i += BDIM) {
            float s = base[i];
            if (s > 0.5f) atomicAdd(&lh[score_bin(s)], 1);
        }
    }
    __syncthreads();

    int* gh = hist + b * NBINS;
    for (int i = tid; i < NBINS; i += BDIM)
        if (lh[i]) atomicAdd(&gh[i], lh[i]);
}

// ---- kernel 3: suffix-scan histogram -> cutoff bin ------------------------
__global__ void scan_kernel(const int* __restrict__ hist, int* __restrict__ cut) {
    if (threadIdx.x != 0) return;
    const int* h = hist + blockIdx.x * NBINS;
    int acc = 0, c = 0;
    for (int i = NBINS - 1; i >= 0; --i) {
        acc += h[i];
        if (acc >= KTOP) { c = i; break; }
    }
    cut[blockIdx.x] = c;
}

// ---- kernel 4: compact candidates at/above cutoff bin (L2-resident pass) --
__global__ __launch_bounds__(BDIM)
void compact_kernel(const float* __restrict__ cls, const int* __restrict__ cut,
                    int* __restrict__ ccount, unsigned long long* __restrict__ cand,
                    int N_, int C_) {
    const int b = blockIdx.y;
    const int total4 = (N_ * C_) >> 2;         // N*C divisible by 4 here
    const int cutb = cut[b];
    const float* base = cls + (size_t)b * (size_t)N_ * (size_t)C_;
    unsigned long long* cb = cand + (size_t)b * CAP;
    int* cnt = ccount + b;

    for (int i4 = blockIdx.x * blockDim.x + threadIdx.x; i4 < total4;
         i4 += gridDim.x * blockDim.x) {
        float4 v = ((const float4*)base)[i4];
        float sv[4] = { v.x, v.y, v.z, v.w };
        #pragma unroll
        for (int j = 0; j < 4; ++j) {
            float s = sv[j];
            if (s > 0.5f && score_bin(s) >= cutb) {
                unsigned int m = (unsigned int)(i4 * 4 + j);  // memory order n*C+c
                unsigned int n = m / (unsigned int)C_;
                unsigned int c = m - n * (unsigned int)C_;
                unsigned int f = c * (unsigned int)N_ + n;    // class-major flat idx
                int pos = atomicAdd(cnt, 1);
                if (pos < CAP)
                    cb[pos] = ((unsigned long long)__float_as_uint(s) << 32) |
                              (unsigned long long)(~f);
            }
        }
    }
}

// ---- kernel 5: per-batch bitonic sort of candidates + gather outputs ------
__global__ __launch_bounds__(SORT_T)
void sort_gather_kernel(const float* __restrict__ boxes, const float* __restrict__ locs,
                        const float* __restrict__ poses,
                        const unsigned long long* __restrict__ cand,
                        const int* __restrict__ ccount,
                        float* __restrict__ oBoxes, float* __restrict__ oLocs,
                        float* __restrict__ oScores, int* __restrict__ oLabels,
                        float* __restrict__ oPoses, int N_) {
    __shared__ unsigned long long keys[CAP];   // 32 KB of the 320 KB WGP LDS
    __shared__ float sScore[KTOP];
    __shared__ int   sBox[KTOP];
    __shared__ int   sLab[KTOP];
    __shared__ int   sValid[KTOP];

    const int b = blockIdx.x;
    const int tid = threadIdx.x;
    int cnt = ccount[b];
    cnt = cnt > CAP ? CAP : cnt;
    const unsigned long long* cb = cand + (size_t)b * CAP;

    for (int i = tid; i < CAP; i += SORT_T)
        keys[i] = (i < cnt) ? cb[i] : 0ULL;    // key 0 == invalid sentinel
    __syncthreads();

    // bitonic sort, descending (score bits high, ~idx low => stable tiebreak)
    for (int k = 2; k <= CAP; k <<= 1) {
        for (int j = k >> 1; j > 0; j >>= 1) {
            for (int i = tid; i < CAP; i += SORT_T) {
                int ixj = i ^ j;
                if (ixj > i) {
                    unsigned long long a = keys[i], c = keys[ixj];
                    bool desc = ((i & k) == 0);
                    if (desc ? (a < c) : (a > c)) { keys[i] = c; keys[ixj] = a; }
                }
            }
            __syncthreads();
        }
    }

    for (int s = tid; s < KTOP; s += SORT_T) {
        unsigned long long key = keys[s];
        float sc = __uint_as_float((unsigned int)(key >> 32));
        int valid = (key != 0ULL) && (sc > 0.5f);
        unsigned int f = ~(unsigned int)(key & 0xFFFFFFFFu);
        sValid[s] = valid;
        sScore[s] = sc;
        sBox[s]   = valid ? (int)(f % (unsigned int)N_) : 0;
        sLab[s]   = valid ? (int)(f / (unsigned int)N_) : -1;
    }
    __syncthreads();

    // 128 "components" per slot: 0-15 boxes, 16-17 locs, 18 score, 19 label, 20-124 poses
    for (int e = tid; e < KTOP * 128; e += SORT_T) {
        int s = e >> 7;
        int c = e & 127;
        int valid = sValid[s];
        int box = sBox[s];
        size_t ob = (size_t)b * KTOP + s;
        if (c < 16)
            oBoxes[ob * 16 + c] = valid ? boxes[((size_t)b * N_ + box) * 16 + c] : -1.0f;
        else if (c < 18)
            oLocs[ob * 2 + (c - 16)] = valid ? locs[((size_t)b * N_ + box) * 2 + (c - 16)] : -1.0f;
        else if (c == 18)
            oScores[ob] = valid ? sScore[s] : -1.0f;
        else if (c == 19)
            oLabels[ob] = sLab[s];
        else if (c < 20 + 105)
            oPoses[ob * 105 + (c - 20)] = valid ? poses[((size_t)b * N_ + box) * 105 + (c - 20)] : -1.0f;
    }
}

// ---- host launcher --------------------------------------------------------
extern "C" void kernel_launch(void* const* d_in, const int* in_sizes, int n_in,
                              void* d_out, int out_size, void* d_ws, size_t ws_size,
                              hipStream_t stream) {
    const float* boxes = (const float*)d_in[0];   // [B,N,16]
    const float* cls   = (const float*)d_in[1];   // [B,N,C]
    const float* locs  = (const float*)d_in[2];   // [B,N,2]
    const float* poses = (const float*)d_in[3];   // [B,N,15,7]

    const int B_ = 8;
    const int N_ = in_sizes[2] / (2 * B_);        // 120000
    const int C_ = in_sizes[1] / (B_ * N_);       // 30
    const int NF = N_ * C_;

    // workspace layout
    int* hist = (int*)d_ws;                       // B*NBINS
    int* cut  = hist + B_ * NBINS;                // B
    int* ccnt = cut + B_;                         // B
    unsigned long long* cand = (unsigned long long*)(ccnt + B_);  // B*CAP (8B aligned)

    // output layout: boxes, locs, scores, labels, poses (flat, in return order)
    float* out = (float*)d_out;
    float* oB  = out;
    float* oL  = oB + (size_t)B_ * KTOP * 16;
    float* oS  = oL + (size_t)B_ * KTOP * 2;
    int*   oLb = (int*)(oS + (size_t)B_ * KTOP);
    float* oP  = (float*)(oLb + (size_t)B_ * KTOP);

    const int nz = B_ * NBINS + 2 * B_;
    zero_ws_kernel<<<(nz + 255) / 256, 256, 0, stream>>>(hist, nz);

    const int tilesPerBatch = NF / (BDIM * 4);
    hist_kernel<<<dim3(HBLK, B_), BDIM, 0, stream>>>(cls, hist, NF, tilesPerBatch);

    scan_kernel<<<B_, 64, 0, stream>>>(hist, cut);

    compact_kernel<<<dim3(CBLK, B_), BDIM, 0, stream>>>(cls, cut, ccnt, cand, N_, C_);

    sort_gather_kernel<<<B_, SORT_T, 0, stream>>>(boxes, locs, poses, cand, ccnt,
                                                  oB, oL, oS, oLb, oP, N_);
}